// AutoODE_COVID_77532749627469
// MI455X (gfx1250) — compile-verified
//
#include <hip/hip_runtime.h>
#include <hip/hip_bf16.h>
#include <math.h>

// ---------------------------------------------------------------------------
// AutoODE COVID SEIRD Euler scan for MI455X (gfx1250).
//   - 4095 sequential steps; per-step global coupling = 2048x2048 f32 matvec.
//   - Matvec done with V_WMMA_F32_16X16X4_F32: A-operand = broadcast x chunk,
//     B-operand = A^T tile, so C[:,n] = flux[rowbase+n] (lane n owns region n).
//   - Persistent 16-block grid + atomic grid barrier per step (scan-limited
//     parallelism; A stays resident in the 192MB L2).
// ---------------------------------------------------------------------------

#define NREG   2048
#define NSTEPS 4096
#define NBLK   16
#define BLKT   256           // 8 waves; each wave covers 16 rows -> 128 rows/block
#define STEPSZ 0.01f

typedef __attribute__((ext_vector_type(2))) float v2f;
typedef __attribute__((ext_vector_type(8))) float v8f;

__global__ __launch_bounds__(256) void seir_init(
    const float* __restrict__ iI, const float* __restrict__ iR,
    const float* __restrict__ iD, const float* __restrict__ iS,
    const float* __restrict__ iE,
    float* __restrict__ out, float* __restrict__ xbuf0, unsigned* bar) {
  int r = blockIdx.x * blockDim.x + threadIdx.x;
  if (r == 0) *bar = 0u;
  if (r < NREG) {
    float I = iI[r], Rr = iR[r], D = iD[r];
    float S = iS[r], E = iE[r];
    size_t o = (size_t)r * (NSTEPS * 3);
    out[o + 0] = I;
    out[o + 1] = Rr;
    out[o + 2] = D;
    xbuf0[r] = (I + E) * S;   // x_0 for the first matvec
  }
}

__global__ __launch_bounds__(BLKT) void seir_scan(
    const float* __restrict__ Amat, const float* __restrict__ beta,
    const float* __restrict__ gamma_, const float* __restrict__ sigma_,
    const float* __restrict__ mu_, const float* __restrict__ a_,
    const float* __restrict__ b_,
    const float* __restrict__ iI, const float* __restrict__ iR,
    const float* __restrict__ iD, const float* __restrict__ iS,
    const float* __restrict__ iE,
    float* __restrict__ out, float* __restrict__ xbuf0,
    float* __restrict__ xbuf1, unsigned* __restrict__ bar) {
  __shared__ float ldsx[NREG];                 // 8 KB staged x-vector

  const int tid  = threadIdx.x;
  const int lane = tid & 31;
  const int wave = tid >> 5;
  const int half = lane >> 4;                  // which 16-lane half of the wave
  const int ln   = lane & 15;
  const int rowbase = blockIdx.x * 128 + wave * 16;
  const int row     = rowbase + ln;            // region owned (lanes 0-15); B-tile row for all

  // B-operand layout (4x16 f32): lane holds 2 consecutive k's of row `row`,
  // half-wave 1 offset by 2 columns.
  const float* Arow = Amat + (size_t)row * NREG + 2 * half;

  // Per-region state + constants (meaningful in lanes 0-15; redundant in 16-31).
  float S = iS[row], E = iE[row], I = iI[row];
  float Rr = iR[row], D = iD[row];
  float gm = gamma_[row], sg = sigma_[row], mv = mu_[row];
  float aa = a_[row], bc = b_[row];

  for (int n = 0; n < NSTEPS - 1; ++n) {
    // ---- grid barrier: all blocks published x_n --------------------------
    __threadfence();                           // make this thread's x writes agent-visible
    __syncthreads();
    if (tid == 0) {
      __hip_atomic_fetch_add(bar, 1u, __ATOMIC_RELEASE, __HIP_MEMORY_SCOPE_AGENT);
      const unsigned target = (unsigned)NBLK * (unsigned)(n + 1);
      while (__hip_atomic_load(bar, __ATOMIC_ACQUIRE, __HIP_MEMORY_SCOPE_AGENT) < target) {
        __builtin_amdgcn_s_sleep(1);
      }
    }
    __syncthreads();

    // ---- stage x_n into LDS (ping-pong: even steps read xbuf0) -----------
    const float4* xr = (const float4*)((n & 1) ? xbuf1 : xbuf0);
    float4* lx4 = (float4*)ldsx;
    for (int i = tid; i < NREG / 4; i += BLKT) lx4[i] = xr[i];
    __syncthreads();

    // ---- flux = A @ x via V_WMMA_F32_16X16X4_F32 -------------------------
    // Aop[m,k] = x[kbase + k] (broadcast over m); per lane: float2 at
    // ldsx[kbase + 2*half]. Bop[k,n] = Amat[rowbase+n][kbase+k]; per lane:
    // float2 at Arow[kbase]. Result: C[m, n] = flux[rowbase+n] for all m.
    v8f c = {0.f, 0.f, 0.f, 0.f, 0.f, 0.f, 0.f, 0.f};
    const float* ax = ldsx + 2 * half;
#pragma unroll 4
    for (int k = 0; k < NREG; k += 4) {
      v2f av = *(const v2f*)(ax + k);
      v2f bv = *(const v2f*)(Arow + k);
      c = __builtin_amdgcn_wmma_f32_16x16x4_f32(
          /*neg_a=*/false, av, /*neg_b=*/false, bv,
          /*c_mod=*/(short)0, c, /*reuse_a=*/false, /*reuse_b=*/false);
    }
    const float flux = c[0];                   // lane ln (<16): flux[rowbase+ln]

    // ---- elementwise SEIRD update (lanes 0-15 own rows) ------------------
    const float bb = beta[(n + 1) & (NREG - 1)];   // torch repeat quirk: beta[(n+1)%R]
    const float tt = (float)(n + 1);
    if (lane < 16) {
      const float ie = I + E;
      const float S1 = S - bb * flux * STEPSZ;
      const float E1 = E + (bb * S * ie - sg * E) * STEPSZ;
      const float I1 = I + (mv * sg * E - gm * I) * STEPSZ;
      const float R1 = Rr + gm * I * STEPSZ;
      const float D1 = D + aa * __expf(-bc * tt * STEPSZ) * (R1 - Rr);

      const size_t o = (size_t)row * (NSTEPS * 3) + (size_t)(n + 1) * 3;
      out[o + 0] = I1;
      out[o + 1] = R1;
      out[o + 2] = D1;

      float* xw = (n & 1) ? xbuf0 : xbuf1;     // publish x_{n+1}
      xw[row] = (I1 + E1) * S1;

      S = S1; E = E1; I = I1; Rr = R1; D = D1;
    }
  }
}

extern "C" void kernel_launch(void* const* d_in, const int* in_sizes, int n_in,
                              void* d_out, int out_size, void* d_ws, size_t ws_size,
                              hipStream_t stream) {
  const float* init_I = (const float*)d_in[0];
  const float* init_R = (const float*)d_in[1];
  const float* init_D = (const float*)d_in[2];
  const float* init_S = (const float*)d_in[3];
  const float* init_E = (const float*)d_in[4];
  const float* Amat   = (const float*)d_in[5];
  const float* beta   = (const float*)d_in[6];
  const float* gamma_ = (const float*)d_in[7];
  const float* sigma_ = (const float*)d_in[8];
  const float* mu_    = (const float*)d_in[9];
  const float* a_     = (const float*)d_in[10];
  const float* b_     = (const float*)d_in[11];
  float* out = (float*)d_out;

  float*    xbuf0 = (float*)d_ws;
  float*    xbuf1 = xbuf0 + NREG;
  unsigned* bar   = (unsigned*)(xbuf1 + NREG);

  seir_init<<<(NREG + 255) / 256, 256, 0, stream>>>(
      init_I, init_R, init_D, init_S, init_E, out, xbuf0, bar);

  seir_scan<<<NBLK, BLKT, 0, stream>>>(
      Amat, beta, gamma_, sigma_, mu_, a_, b_,
      init_I, init_R, init_D, init_S, init_E,
      out, xbuf0, xbuf1, bar);
}